// Discriminator_48730698940787
// MI455X (gfx1250) — compile-verified
//
#include <hip/hip_runtime.h>
#include <math.h>

#define NWIN   3
#define CIN    256
#define C2     512
#define HW     96
#define PLANE  (HW*HW)      // 9216
#define OD0    94
#define L0     (OD0*OD0)    // 8836
#define L1     4
#define NBATCH 2
#define NROWS  6            // row = n*3 + w
#define NT0    ((L0 + 15)/16) // 553 column tiles

// workspace layout (float offsets)
#define WS_XF0 0
#define WS_XF1 (WS_XF0 + NBATCH*C2*L0)     // 2*512*8836
#define WS_XF2 (WS_XF1 + NBATCH*C2*L1)
#define WS_CEN (WS_XF2 + NBATCH*C2)
#define WS_U   (WS_CEN + NBATCH*3*C2)
#define WS_T   (WS_U + NWIN*NROWS*C2)
#define WS_UP  (WS_T + 32)                 // zero-padded win0 A: [n][16][C2]
#define WS_PW0 (WS_UP + NBATCH*16*C2)
#define WS_AGG (WS_PW0 + NROWS*L0)

typedef __attribute__((ext_vector_type(2))) float v2f;
typedef __attribute__((ext_vector_type(8))) float v8f;

__device__ __forceinline__ float blk_sum(float v, float* red) {
    int tid = threadIdx.x;
    red[tid] = v; __syncthreads();
    for (int off = 128; off > 0; off >>= 1) {
        if (tid < off) red[tid] += red[tid + off];
        __syncthreads();
    }
    float r = red[0]; __syncthreads();
    return r;
}

// ---------- window stats via LDS plane: box sums for K=3/50/96 ----------
__global__ __launch_bounds__(256) void k_stats(const float* __restrict__ feature,
                                               float* __restrict__ ws) {
    __shared__ float sf[PLANE];
    __shared__ float red[256];
    int tid = threadIdx.x;
    int n = blockIdx.x / CIN, c = blockIdx.x % CIN;
    const float* src = feature + (size_t)(n*CIN + c)*PLANE;
    for (int i = tid; i < PLANE; i += 256) sf[i] = src[i];
    __syncthreads();
    float* xf0 = ws + WS_XF0;
    float* xf1 = ws + WS_XF1;
    float* xf2 = ws + WS_XF2;
    float* cen = ws + WS_CEN;

    // K=96 (single window == whole plane)
    float s = 0.f, s2 = 0.f;
    for (int i = tid; i < PLANE; i += 256) { float v = sf[i]; s += v; s2 += v*v; }
    float S = blk_sum(s, red), S2 = blk_sum(s2, red);
    if (tid == 0) {
        float m = S * (1.f/9216.f);
        float var = fmaxf(S2 * (1.f/9216.f) - m*m, 0.f);
        float sd = sqrtf(var + 1e-12f);
        xf2[n*C2 + c]        = m;
        xf2[n*C2 + c + CIN]  = sd;
        cen[(n*3 + 2)*C2 + c]       = m;
        cen[(n*3 + 2)*C2 + c + CIN] = sd;
    }

    // K=50, stride 24 -> 2x2 windows at y0,x0 in {0,24}
    float c1m = 0.f, c1s = 0.f;
    for (int wy = 0; wy < 2; wy++) {
        for (int wx = 0; wx < 2; wx++) {
            int y0 = wy*24, x0 = wx*24;
            float p = 0.f, p2 = 0.f;
            for (int i = tid; i < 2500; i += 256) {
                int yy = i / 50, xx = i % 50;
                float v = sf[(y0+yy)*HW + x0 + xx];
                p += v; p2 += v*v;
            }
            float P = blk_sum(p, red), P2 = blk_sum(p2, red);
            if (tid == 0) {
                float m = P * (1.f/2500.f);
                float var = fmaxf(P2 * (1.f/2500.f) - m*m, 0.f);
                float sd = sqrtf(var + 1e-12f);
                int l = wy*2 + wx;
                xf1[(n*C2 + c)*L1 + l]       = m;
                xf1[(n*C2 + c + CIN)*L1 + l] = sd;
                c1m += m; c1s += sd;
            }
        }
    }
    if (tid == 0) {
        cen[(n*3 + 1)*C2 + c]       = c1m * 0.25f;
        cen[(n*3 + 1)*C2 + c + CIN] = c1s * 0.25f;
    }

    // K=3, stride 1 -> 94x94 windows
    float am = 0.f, as = 0.f;
    for (int p = tid; p < L0; p += 256) {
        int y = p / OD0, x = p % OD0;
        float t = 0.f, t2 = 0.f;
        #pragma unroll
        for (int dy = 0; dy < 3; dy++) {
            const float* r = &sf[(y+dy)*HW + x];
            float a = r[0], b = r[1], d = r[2];
            t  += a + b + d;
            t2 += a*a + b*b + d*d;
        }
        float m = t * (1.f/9.f);
        float var = fmaxf(t2 * (1.f/9.f) - m*m, 0.f);
        float sd = sqrtf(var + 1e-12f);
        xf0[(size_t)(n*C2 + c)*L0 + p]       = m;
        xf0[(size_t)(n*C2 + c + CIN)*L0 + p] = sd;
        am += m; as += sd;
    }
    float AM = blk_sum(am, red), AS = blk_sum(as, red);
    if (tid == 0) {
        cen[(n*3 + 0)*C2 + c]       = AM / (float)L0;
        cen[(n*3 + 0)*C2 + c + CIN] = AS / (float)L0;
    }
}

// ---------- theta_x = centers@theta_w^T + b ; u = theta_x@phi_w ; t = theta_x.phi_b ----------
// Also emits Upad: window-0 u rows zero-padded to a 16x512 WMMA A-matrix per batch.
__global__ __launch_bounds__(256) void k_theta_u(const float* __restrict__ theta_w,
                                                 const float* __restrict__ theta_b,
                                                 const float* __restrict__ phi_w,
                                                 const float* __restrict__ phi_b,
                                                 float* __restrict__ ws) {
    __shared__ float th[NROWS*CIN];
    int tid = threadIdx.x;
    const float* cen = ws + WS_CEN;
    for (int o = tid; o < NROWS*CIN; o += 256) {
        int row = o / CIN, j = o % CIN;
        const float* cv = cen + row*C2;
        const float* wv = theta_w + (size_t)j*C2;
        float acc = theta_b[j];
        for (int k = 0; k < C2; k++) acc += cv[k]*wv[k];
        th[row*CIN + j] = acc;
    }
    __syncthreads();
    float* u  = ws + WS_U;
    float* tt = ws + WS_T;
    float* up = ws + WS_UP;
    for (int o = tid; o < NWIN*NROWS*C2; o += 256) {
        int win = o / (NROWS*C2), rem = o % (NROWS*C2);
        int row = rem / C2, cch = rem % C2;
        const float* tv = th + row*CIN;
        const float* pw = phi_w + (size_t)win*CIN*C2 + cch;
        float acc = 0.f;
        for (int j = 0; j < CIN; j++) acc += tv[j] * pw[(size_t)j*C2];
        u[(win*NROWS + row)*C2 + cch] = acc;
        if (win == 0) {
            int n = row / 3, w = row % 3;
            up[(n*16 + w)*C2 + cch] = acc;   // rows 0..2 of the padded A
        }
    }
    // zero the padding rows 3..15 of Upad
    for (int o = tid; o < NBATCH*13*C2; o += 256) {
        int n = o / (13*C2), rem = o % (13*C2);
        int r = 3 + rem / C2, cch = rem % C2;
        up[(n*16 + r)*C2 + cch] = 0.f;
    }
    if (tid < NWIN*NROWS) {
        int win = tid / NROWS, row = tid % NROWS;
        const float* tv = th + row*CIN;
        const float* pb = phi_b + win*CIN;
        float acc = 0.f;
        for (int j = 0; j < CIN; j++) acc += tv[j]*pb[j];
        tt[tid] = acc;
    }
}

// ---------- window-0 logits: (3x512 padded to 16) @ (512 x 8836) via V_WMMA_F32_16X16X4_F32 ----------
// Upad removes all predication from the hot loop: per k-step = 1 b64 load (A) + 2 b32 loads (B) + 1 wmma.
__global__ __launch_bounds__(32) void k_logits0(float* __restrict__ ws) {
    int lane = threadIdx.x;
    int n    = blockIdx.x / NT0;
    int tile = blockIdx.x % NT0;
    int l0   = tile * 16;
    const float* X  = ws + WS_XF0 + (size_t)n*C2*L0;
    const float* UP = ws + WS_UP + n*16*C2;
    const float* T  = ws + WS_T;
    float* out = ws + WS_PW0 + n*3*L0;

    int half = lane >> 4;        // lanes 16-31 hold K+2,K+3
    int m    = lane & 15;        // A: M row / B: N column
    int col  = l0 + m; if (col >= L0) col = L0 - 1;   // clamp; extra cols never stored
    const float* arow = UP + m*C2 + half*2;           // &Upad[m][half*2]
    const float* bcol = X + col + (size_t)half*2*L0;  // &X[half*2][col]

    v8f acc = {0.f,0.f,0.f,0.f,0.f,0.f,0.f,0.f};
    #pragma unroll 4
    for (int k = 0; k < C2; k += 4) {
        v2f a = *(const v2f*)(arow + k);   // A[m][ka], A[m][ka+1] (8B aligned)
        v2f b;
        b.x = bcol[(size_t)k*L0];
        b.y = bcol[(size_t)(k+1)*L0];
        acc = __builtin_amdgcn_wmma_f32_16x16x4_f32(
                  false, a, false, b, (short)0, acc, false, false);
    }
    // D rows 0..2 live in acc[0..2] on lanes 0-15
    if (lane < 16 && (l0 + lane) < L0) {
        for (int w = 0; w < 3; w++)
            out[w*L0 + l0 + lane] = (acc[w] + T[n*3 + w]) * 0.0625f; // /sqrt(256)
    }
}

// ---------- softmax over L0 per (n,w) row, in place ----------
__global__ __launch_bounds__(256) void k_softmax0(float* __restrict__ ws) {
    __shared__ float red[256];
    int tid = threadIdx.x;
    float* p = ws + WS_PW0 + (size_t)blockIdx.x * L0;
    float mx = -1e30f;
    for (int l = tid; l < L0; l += 256) mx = fmaxf(mx, p[l]);
    red[tid] = mx; __syncthreads();
    for (int off = 128; off > 0; off >>= 1) {
        if (tid < off) red[tid] = fmaxf(red[tid], red[tid+off]);
        __syncthreads();
    }
    mx = red[0]; __syncthreads();
    float s = 0.f;
    for (int l = tid; l < L0; l += 256) s += expf(p[l] - mx);
    red[tid] = s; __syncthreads();
    for (int off = 128; off > 0; off >>= 1) {
        if (tid < off) red[tid] += red[tid+off];
        __syncthreads();
    }
    float inv = 1.f / red[0];
    for (int l = tid; l < L0; l += 256) p[l] = expf(p[l] - mx) * inv;
}

// ---------- agg0[n,w,c] = sum_l pw*xf - centers (softmax rows sum to 1) ----------
__global__ __launch_bounds__(256) void k_agg0(float* __restrict__ ws) {
    __shared__ float red0[256], red1[256], red2[256];
    int tid = threadIdx.x;
    int n = blockIdx.x >> 9, ch = blockIdx.x & 511;
    const float* x = ws + WS_XF0 + (size_t)(n*C2 + ch)*L0;
    const float* p = ws + WS_PW0 + n*3*L0;
    float s0 = 0.f, s1 = 0.f, s2 = 0.f;
    for (int l = tid; l < L0; l += 256) {
        float v = x[l];
        s0 += p[l]*v; s1 += p[L0 + l]*v; s2 += p[2*L0 + l]*v;
    }
    red0[tid] = s0; red1[tid] = s1; red2[tid] = s2; __syncthreads();
    for (int off = 128; off > 0; off >>= 1) {
        if (tid < off) {
            red0[tid] += red0[tid+off];
            red1[tid] += red1[tid+off];
            red2[tid] += red2[tid+off];
        }
        __syncthreads();
    }
    if (tid < 3) {
        float sv = (tid == 0) ? red0[0] : (tid == 1) ? red1[0] : red2[0];
        float* agg = ws + WS_AGG;
        const float* cen = ws + WS_CEN;
        agg[(n*3 + tid)*C2 + ch] = sv - cen[(n*3 + tid)*C2 + ch];
    }
}

// ---------- windows 1 (L=4) and 2 (L=1): logits+softmax+agg in one tiny block ----------
__global__ __launch_bounds__(256) void k_small(float* __restrict__ ws) {
    __shared__ float lg[NROWS*L1];
    int tid = threadIdx.x;
    const float* xf1 = ws + WS_XF1;
    const float* xf2 = ws + WS_XF2;
    const float* cen = ws + WS_CEN;
    const float* u   = ws + WS_U;
    const float* T   = ws + WS_T;
    float* agg = ws + WS_AGG;
    if (tid < NROWS*L1) {
        int row = tid / L1, l = tid % L1;
        int n = row / 3;
        const float* uv = u + (NROWS + row)*C2;    // window 1
        float acc = 0.f;
        for (int c = 0; c < C2; c++) acc += uv[c] * xf1[(n*C2 + c)*L1 + l];
        lg[tid] = (acc + T[NROWS + row]) * 0.0625f;
    }
    __syncthreads();
    if (tid < NROWS) {
        float mx = lg[tid*L1];
        for (int l = 1; l < L1; l++) mx = fmaxf(mx, lg[tid*L1 + l]);
        float s = 0.f;
        for (int l = 0; l < L1; l++) { float e = expf(lg[tid*L1 + l] - mx); lg[tid*L1 + l] = e; s += e; }
        float inv = 1.f / s;
        for (int l = 0; l < L1; l++) lg[tid*L1 + l] *= inv;
    }
    __syncthreads();
    for (int o = tid; o < NROWS*C2; o += 256) {
        int row = o / C2, c = o % C2;
        int n = row / 3;
        float a1 = 0.f;
        for (int l = 0; l < L1; l++) a1 += lg[row*L1 + l] * xf1[(n*C2 + c)*L1 + l];
        agg[(NROWS + row)*C2 + c]   = a1 - cen[row*C2 + c];                 // window 1
        agg[(2*NROWS + row)*C2 + c] = xf2[n*C2 + c] - cen[row*C2 + c];      // window 2 (pw == 1)
    }
}

// ---------- normalize + per-window MLP + BCE-with-logits mean -> scalar ----------
__global__ __launch_bounds__(256) void k_mlp(const float* __restrict__ m1w, const float* __restrict__ m1b,
                                             const float* __restrict__ m2w, const float* __restrict__ m2b,
                                             const float* __restrict__ m3w, const float* __restrict__ m3b,
                                             const float* __restrict__ m4w, const float* __restrict__ m4b,
                                             const int* __restrict__ label,
                                             float* __restrict__ ws, float* __restrict__ out) {
    __shared__ float vin[C2];
    __shared__ float h1[CIN];
    __shared__ float h2[256];
    __shared__ float h3[128];
    __shared__ float red[256];
    int tid = threadIdx.x;
    float lf = (float)label[0];
    const float* agg = ws + WS_AGG;
    float total = 0.f;
    for (int win = 0; win < NWIN; win++) {
        float wsum = 0.f;
        for (int row = 0; row < NROWS; row++) {
            __syncthreads();
            float ss = 0.f;
            for (int c = tid; c < C2; c += 256) {
                float v = agg[(win*NROWS + row)*C2 + c];
                vin[c] = v; ss += v*v;
            }
            red[tid] = ss; __syncthreads();
            for (int off = 128; off > 0; off >>= 1) { if (tid < off) red[tid] += red[tid+off]; __syncthreads(); }
            float scl = 1.f / fmaxf(sqrtf(red[0]), 1e-12f);
            __syncthreads();
            {   // layer1: 512 -> 256, leaky relu 0.2
                const float* w = m1w + ((size_t)win*CIN + tid)*C2;
                float acc = m1b[win*CIN + tid];
                for (int c = 0; c < C2; c++) acc += vin[c]*scl*w[c];
                h1[tid] = (acc > 0.f) ? acc : 0.2f*acc;
            }
            __syncthreads();
            {   // layer2: 256 -> 256
                const float* w = m2w + ((size_t)win*256 + tid)*CIN;
                float acc = m2b[win*256 + tid];
                for (int c = 0; c < CIN; c++) acc += h1[c]*w[c];
                h2[tid] = (acc > 0.f) ? acc : 0.2f*acc;
            }
            __syncthreads();
            if (tid < 128) {   // layer3: 256 -> 128
                const float* w = m3w + ((size_t)win*128 + tid)*256;
                float acc = m3b[win*128 + tid];
                for (int c = 0; c < 256; c++) acc += h2[c]*w[c];
                h3[tid] = (acc > 0.f) ? acc : 0.2f*acc;
            }
            __syncthreads();
            float part = (tid < 128) ? h3[tid]*m4w[win*128 + tid] : 0.f;
            red[tid] = part; __syncthreads();
            for (int off = 128; off > 0; off >>= 1) { if (tid < off) red[tid] += red[tid+off]; __syncthreads(); }
            if (tid == 0) {
                float logit = red[0] + m4b[win];
                float sp = fmaxf(logit, 0.f) + log1pf(expf(-fabsf(logit)));  // stable softplus
                wsum += sp - logit*lf;
            }
            __syncthreads();
        }
        if (tid == 0) total += wsum / (float)NROWS;
    }
    if (tid == 0) out[0] = total;
}

extern "C" void kernel_launch(void* const* d_in, const int* in_sizes, int n_in,
                              void* d_out, int out_size, void* d_ws, size_t ws_size,
                              hipStream_t stream) {
    (void)in_sizes; (void)n_in; (void)out_size; (void)ws_size;
    const float* feature = (const float*)d_in[0];
    const float* theta_w = (const float*)d_in[1];
    const float* theta_b = (const float*)d_in[2];
    const float* phi_w   = (const float*)d_in[3];
    const float* phi_b   = (const float*)d_in[4];
    const float* m1w = (const float*)d_in[5];
    const float* m1b = (const float*)d_in[6];
    const float* m2w = (const float*)d_in[7];
    const float* m2b = (const float*)d_in[8];
    const float* m3w = (const float*)d_in[9];
    const float* m3b = (const float*)d_in[10];
    const float* m4w = (const float*)d_in[11];
    const float* m4b = (const float*)d_in[12];
    const int*   label = (const int*)d_in[13];
    float* ws  = (float*)d_ws;
    float* out = (float*)d_out;

    k_stats   <<<dim3(NBATCH*CIN), dim3(256), 0, stream>>>(feature, ws);
    k_theta_u <<<dim3(1),          dim3(256), 0, stream>>>(theta_w, theta_b, phi_w, phi_b, ws);
    k_logits0 <<<dim3(NBATCH*NT0), dim3(32),  0, stream>>>(ws);
    k_softmax0<<<dim3(NROWS),      dim3(256), 0, stream>>>(ws);
    k_agg0    <<<dim3(NBATCH*C2),  dim3(256), 0, stream>>>(ws);
    k_small   <<<dim3(1),          dim3(256), 0, stream>>>(ws);
    k_mlp     <<<dim3(1),          dim3(256), 0, stream>>>(m1w,m1b,m2w,m2b,m3w,m3b,m4w,m4b,label,ws,out);
}